// ContactNet_69380901699957
// MI455X (gfx1250) — compile-verified
//
#include <hip/hip_runtime.h>

#define DEV __device__ __forceinline__

typedef __attribute__((ext_vector_type(2))) float v2f;
typedef __attribute__((ext_vector_type(8))) float v8f;

constexpr int B_ = 8;
constexpr int N_ = 4096;
constexpr float EPS_BN = 1e-5f;

// ---------------------------------------------------------------------------
// Workspace arena (offsets in floats)
// ---------------------------------------------------------------------------
constexpr size_t OF_L0XYZ = 0;                     // 8*4096*3
constexpr size_t OF_FIDX1 = OF_L0XYZ + 98304;      // int 8*256
constexpr size_t OF_L1XYZ = OF_FIDX1 + 2048;       // 8*256*3
constexpr size_t OF_GI1   = OF_L1XYZ + 6144;       // int 8*256*32
constexpr size_t OF_FIDX2 = OF_GI1 + 65536;        // int 8*128
constexpr size_t OF_L2XYZ = OF_FIDX2 + 1024;       // 8*128*3
constexpr size_t OF_GI2   = OF_L2XYZ + 3072;       // int 8*128*64
constexpr size_t OF_L1PTS = OF_GI2 + 65536;        // 8*256*128
constexpr size_t OF_L2PTS = OF_L1PTS + 262144;     // 8*128*256
constexpr size_t OF_XENC  = OF_L2PTS + 262144;     // 8*512
constexpr size_t OF_FC4IN = OF_XENC + 4096;        // 8*592
constexpr size_t OF_L3    = OF_FC4IN + 4736;       // 8*512
constexpr size_t OF_L2D   = OF_L3 + 4096;          // 8*128*256
constexpr size_t OF_L1D   = OF_L2D + 262144;       // 8*256*128
constexpr size_t OF_IDX2  = OF_L1D + 262144;       // int 8*256*3
constexpr size_t OF_W2    = OF_IDX2 + 6144;        // 8*256*3
constexpr size_t OF_IDX1  = OF_W2 + 6144;          // int 8*4096*3
constexpr size_t OF_W1    = OF_IDX1 + 98304;       // 8*4096*3
constexpr size_t OF_MEAN  = OF_W1 + 98304;         // 1024
constexpr size_t OF_VAR   = OF_MEAN + 1024;        // 1024
constexpr size_t OF_G3    = OF_VAR + 1024;         // 1024*259
constexpr size_t OF_FP3IN = OF_G3 + 265216;        // 1024*768
constexpr size_t OF_FP2IN = OF_FP3IN + 786432;     // 2048*384
constexpr size_t OF_FP1IN = OF_FP2IN + 786432;     // 32768*137
constexpr size_t OF_CL    = OF_FP1IN + 4489216;    // 32768
constexpr size_t OF_BIGA  = OF_CL + 32768;         // 8.4M  (h1a,h2a,h3a,f3a,f2a,f1a,feat)
constexpr size_t OF_BIGB  = OF_BIGA + 8388608;     // 16.8M (h1b,h2b,h3b,l0d)
constexpr size_t OF_XO    = OF_BIGB + 16777216;    // 32768*778
// total ~58.5M floats ~234MB

// ---------------------------------------------------------------------------
// Misc prep kernels
// ---------------------------------------------------------------------------
__global__ void build_l0xyz_kernel(const float* __restrict__ verts, float* __restrict__ xyz) {
    int i = blockIdx.x * blockDim.x + threadIdx.x;       // over B*N
    if (i >= B_ * N_) return;
    int b = i / N_, n = i - b * N_;
    xyz[(size_t)i * 3 + 0] = verts[((size_t)b * 3 + 0) * N_ + n];
    xyz[(size_t)i * 3 + 1] = verts[((size_t)b * 3 + 1) * N_ + n];
    xyz[(size_t)i * 3 + 2] = verts[((size_t)b * 3 + 2) * N_ + n];
}

// Farthest point sampling: one block per batch, sequential npoint iterations.
__global__ void fps_kernel(const float* __restrict__ xyz, int* __restrict__ fidx,
                           int Npts, int np) {
    __shared__ float dist[4096];
    __shared__ float bestd[512];
    __shared__ int   besti[512];
    __shared__ int   far_s;
    int b = blockIdx.x, tid = threadIdx.x;
    const float* base = xyz + (size_t)b * Npts * 3;
    for (int j = tid; j < Npts; j += 512) dist[j] = 1e10f;
    if (tid == 0) far_s = 0;
    __syncthreads();
    for (int it = 0; it < np; ++it) {
        int f = far_s;
        if (tid == 0) fidx[b * np + it] = f;
        float cx = base[f * 3 + 0], cy = base[f * 3 + 1], cz = base[f * 3 + 2];
        float bd = -1.0f; int bi = 0;
        for (int j = tid; j < Npts; j += 512) {
            float dx = base[j * 3 + 0] - cx;
            float dy = base[j * 3 + 1] - cy;
            float dz = base[j * 3 + 2] - cz;
            float d = dx * dx + dy * dy + dz * dz;
            float nd = fminf(dist[j], d);
            dist[j] = nd;
            if (nd > bd) { bd = nd; bi = j; }
        }
        bestd[tid] = bd; besti[tid] = bi;
        __syncthreads();
        for (int s = 256; s > 0; s >>= 1) {
            if (tid < s) {
                if (bestd[tid + s] > bestd[tid] ||
                    (bestd[tid + s] == bestd[tid] && besti[tid + s] < besti[tid])) {
                    bestd[tid] = bestd[tid + s];
                    besti[tid] = besti[tid + s];
                }
            }
            __syncthreads();
        }
        if (tid == 0) far_s = besti[0];
        __syncthreads();
    }
}

__global__ void gather_xyz_kernel(const float* __restrict__ xyz, const int* __restrict__ fidx,
                                  float* __restrict__ out, int Npts, int np) {
    int i = blockIdx.x * blockDim.x + threadIdx.x;       // over B*np
    if (i >= B_ * np) return;
    int b = i / np;
    int j = fidx[i];
    out[(size_t)i * 3 + 0] = xyz[((size_t)b * Npts + j) * 3 + 0];
    out[(size_t)i * 3 + 1] = xyz[((size_t)b * Npts + j) * 3 + 1];
    out[(size_t)i * 3 + 2] = xyz[((size_t)b * Npts + j) * 3 + 2];
}

// Ball query: first nsample indices (ascending) with d<=r2, padded with first hit.
__global__ void ball_kernel(const float* __restrict__ xyz, const float* __restrict__ nxyz,
                            int* __restrict__ gi, int Npts, int np, int ns, float r2) {
    int i = blockIdx.x * blockDim.x + threadIdx.x;       // over B*np
    if (i >= B_ * np) return;
    int b = i / np;
    const float* base = xyz + (size_t)b * Npts * 3;
    float cx = nxyz[(size_t)i * 3 + 0], cy = nxyz[(size_t)i * 3 + 1], cz = nxyz[(size_t)i * 3 + 2];
    int* out = gi + (size_t)i * ns;
    int cnt = 0, first = 0;
    bool got = false;
    for (int j = 0; j < Npts && cnt < ns; ++j) {
        float dx = base[j * 3 + 0] - cx;
        float dy = base[j * 3 + 1] - cy;
        float dz = base[j * 3 + 2] - cz;
        float d = dx * dx + dy * dy + dz * dz;
        if (d <= r2) {
            if (!got) { first = j; got = true; }
            out[cnt++] = j;
        }
    }
    for (; cnt < ns; ++cnt) out[cnt] = first;
}

// 3-nearest neighbors + normalized inverse-distance weights
__global__ void knn3_kernel(const float* __restrict__ xyz1, const float* __restrict__ xyz2,
                            int n1, int n2, int* __restrict__ idx, float* __restrict__ w) {
    int i = blockIdx.x * blockDim.x + threadIdx.x;       // over B*n1
    if (i >= B_ * n1) return;
    int b = i / n1;
    float x = xyz1[(size_t)i * 3 + 0], y = xyz1[(size_t)i * 3 + 1], z = xyz1[(size_t)i * 3 + 2];
    const float* base = xyz2 + (size_t)b * n2 * 3;
    float d0 = 3.4e38f, d1 = 3.4e38f, d2 = 3.4e38f;
    int   i0 = 0, i1 = 0, i2 = 0;
    for (int j = 0; j < n2; ++j) {
        float dx = base[j * 3 + 0] - x;
        float dy = base[j * 3 + 1] - y;
        float dz = base[j * 3 + 2] - z;
        float d = dx * dx + dy * dy + dz * dz;
        if (d < d0)      { d2 = d1; i2 = i1; d1 = d0; i1 = i0; d0 = d; i0 = j; }
        else if (d < d1) { d2 = d1; i2 = i1; d1 = d;  i1 = j; }
        else if (d < d2) { d2 = d;  i2 = j; }
    }
    float w0 = 1.0f / (d0 + 1e-8f), w1 = 1.0f / (d1 + 1e-8f), w2 = 1.0f / (d2 + 1e-8f);
    float ws = w0 + w1 + w2;
    idx[(size_t)i * 3 + 0] = i0; idx[(size_t)i * 3 + 1] = i1; idx[(size_t)i * 3 + 2] = i2;
    w[(size_t)i * 3 + 0] = w0 / ws; w[(size_t)i * 3 + 1] = w1 / ws; w[(size_t)i * 3 + 2] = w2 / ws;
}

// ---------------------------------------------------------------------------
// WMMA GEMM kernels: f32 16x16x4, 32x32 C macro-tile per wave (4 accumulators).
// Out-of-range rows/cols are CLAMPED (loads stay in-bounds; WMMA routes A row r
// only to C row r and B col c only to C col c, and those C slots are never
// stored), so the main K loop has zero per-element guards.
// ---------------------------------------------------------------------------
__global__ void gemm_bias_kernel(const float* __restrict__ A, const float* __restrict__ W,
                                 const float* __restrict__ bias, float* __restrict__ Y,
                                 int M, int K, int Nc) {
    int wave  = threadIdx.x >> 5;
    int tileM = blockIdx.y * 4 + wave;                    // 32 rows per tile
    if (tileM >= (M + 31) / 32) return;                   // wave-uniform
    int lane = threadIdx.x & 31;
    int half = lane >> 4;
    int lid  = lane & 15;
    int n0 = blockIdx.x * 32 + lid, n1 = n0 + 16;
    int nc0 = min(n0, Nc - 1),      nc1 = min(n1, Nc - 1);
    int m0 = tileM * 32 + lid,      m1 = m0 + 16;
    int mc0 = min(m0, M - 1),       mc1 = min(m1, M - 1);

    const float* Ar0 = A + (size_t)mc0 * K;
    const float* Ar1 = A + (size_t)mc1 * K;
    const float* Wc0 = W + nc0;
    const float* Wc1 = W + nc1;

    v8f acc00 = {}, acc01 = {}, acc10 = {}, acc11 = {};
    auto wstep = [&](v2f a0, v2f a1, v2f bb0, v2f bb1) {
        acc00 = __builtin_amdgcn_wmma_f32_16x16x4_f32(false, a0, false, bb0, (short)0, acc00, false, false);
        acc01 = __builtin_amdgcn_wmma_f32_16x16x4_f32(false, a0, false, bb1, (short)0, acc01, false, false);
        acc10 = __builtin_amdgcn_wmma_f32_16x16x4_f32(false, a1, false, bb0, (short)0, acc10, false, false);
        acc11 = __builtin_amdgcn_wmma_f32_16x16x4_f32(false, a1, false, bb1, (short)0, acc11, false, false);
    };

    int K4 = K & ~3;
    if ((K & 1) == 0) {
        // A elements (k,k+1) are contiguous and 8B-aligned: vector loads
        for (int kb = 0; kb < K4; kb += 4) {
            int ka = kb + 2 * half;
            __builtin_prefetch(Ar0 + ka + 64, 0, 1);
            v2f a0 = *(const v2f*)(Ar0 + ka);
            v2f a1 = *(const v2f*)(Ar1 + ka);
            v2f bb0 = { Wc0[(size_t)ka * Nc], Wc0[(size_t)(ka + 1) * Nc] };
            v2f bb1 = { Wc1[(size_t)ka * Nc], Wc1[(size_t)(ka + 1) * Nc] };
            wstep(a0, a1, bb0, bb1);
        }
    } else {
        for (int kb = 0; kb < K4; kb += 4) {
            int ka = kb + 2 * half;
            v2f a0 = { Ar0[ka], Ar0[ka + 1] };
            v2f a1 = { Ar1[ka], Ar1[ka + 1] };
            v2f bb0 = { Wc0[(size_t)ka * Nc], Wc0[(size_t)(ka + 1) * Nc] };
            v2f bb1 = { Wc1[(size_t)ka * Nc], Wc1[(size_t)(ka + 1) * Nc] };
            wstep(a0, a1, bb0, bb1);
        }
    }
    for (int kb = K4; kb < K; kb += 4) {                  // single guarded tail
        int ka = kb + 2 * half;
        v2f a0 = { (ka < K) ? Ar0[ka] : 0.f, (ka + 1 < K) ? Ar0[ka + 1] : 0.f };
        v2f a1 = { (ka < K) ? Ar1[ka] : 0.f, (ka + 1 < K) ? Ar1[ka + 1] : 0.f };
        v2f bb0 = { (ka < K) ? Wc0[(size_t)ka * Nc] : 0.f,
                    (ka + 1 < K) ? Wc0[(size_t)(ka + 1) * Nc] : 0.f };
        v2f bb1 = { (ka < K) ? Wc1[(size_t)ka * Nc] : 0.f,
                    (ka + 1 < K) ? Wc1[(size_t)(ka + 1) * Nc] : 0.f };
        wstep(a0, a1, bb0, bb1);
    }

    float bi0 = bias[nc0], bi1 = bias[nc1];
    bool full = (tileM * 32 + 31 < M) && (blockIdx.x * 32 + 31 < Nc);
    if (full) {
        #pragma unroll
        for (int i = 0; i < 8; ++i) {
            int r0 = tileM * 32 + i + 8 * half, r1 = r0 + 16;
            Y[(size_t)r0 * Nc + n0] = acc00[i] + bi0;
            Y[(size_t)r0 * Nc + n1] = acc01[i] + bi1;
            Y[(size_t)r1 * Nc + n0] = acc10[i] + bi0;
            Y[(size_t)r1 * Nc + n1] = acc11[i] + bi1;
        }
    } else {
        #pragma unroll
        for (int i = 0; i < 8; ++i) {
            int r0 = tileM * 32 + i + 8 * half, r1 = r0 + 16;
            if (r0 < M && n0 < Nc) Y[(size_t)r0 * Nc + n0] = acc00[i] + bi0;
            if (r0 < M && n1 < Nc) Y[(size_t)r0 * Nc + n1] = acc01[i] + bi1;
            if (r1 < M && n0 < Nc) Y[(size_t)r1 * Nc + n0] = acc10[i] + bi0;
            if (r1 < M && n1 < Nc) Y[(size_t)r1 * Nc + n1] = acc11[i] + bi1;
        }
    }
}

// Guarded feature fetch for SA gather-GEMM (prologue / tail only)
// LEVEL 1: p0=feat_object (B,6,N), p1=contacts (B,778,N), p2=contactness (B,1,N)
// LEVEL 2: lp = l1_pts row base (128 ch)
template <int LEVEL>
DEV float sa_fetch(int c, int K, int bb, int jj,
                   const float* p0, const float* p1, const float* p2,
                   const float* lp, float d0, float d1, float d2) {
    if (c >= K) return 0.f;
    if (c == 0) return d0;
    if (c == 1) return d1;
    if (c == 2) return d2;
    if (LEVEL == 1) {
        if (c < 9)   return p0[((size_t)bb * 6 + (c - 3)) * N_ + jj];
        if (c < 787) return p1[((size_t)bb * 778 + (c - 9)) * N_ + jj];
        return p2[(size_t)bb * N_ + jj];
    } else {
        return lp[c - 3];
    }
}

// SA gather-GEMM: A rows are grouped points, features fetched on the fly.
// K loop is segmented: guarded prologue (dxyz + small feats), branch-free
// uniform middle, guarded tail. M and Nc are multiples of 32 here.
template <int LEVEL>
__global__ void sa_gemm_kernel(const float* __restrict__ xyz, const float* __restrict__ nxyz,
                               const int* __restrict__ gi,
                               const float* __restrict__ p0, const float* __restrict__ p1,
                               const float* __restrict__ p2,
                               const float* __restrict__ W, const float* __restrict__ bias,
                               float* __restrict__ Y,
                               int Npts, int npoint, int nsample, int K, int Nc) {
    int wave  = threadIdx.x >> 5;
    int tileM = blockIdx.y * 4 + wave;                    // 32 rows per tile
    int lane = threadIdx.x & 31;
    int half = lane >> 4;
    int lid  = lane & 15;
    int n0 = blockIdx.x * 32 + lid, n1 = n0 + 16;
    int ps = npoint * nsample;

    int m0 = tileM * 32 + lid, m1 = m0 + 16;
    int b0 = m0 / ps; int r0 = m0 - b0 * ps; int s0 = r0 / nsample; int q0 = r0 - s0 * nsample;
    int b1 = m1 / ps; int r1 = m1 - b1 * ps; int s1 = r1 / nsample; int q1 = r1 - s1 * nsample;
    int j0 = gi[((size_t)b0 * npoint + s0) * nsample + q0];
    int j1 = gi[((size_t)b1 * npoint + s1) * nsample + q1];

    float dA0 = xyz[((size_t)b0 * Npts + j0) * 3 + 0] - nxyz[((size_t)b0 * npoint + s0) * 3 + 0];
    float dA1 = xyz[((size_t)b0 * Npts + j0) * 3 + 1] - nxyz[((size_t)b0 * npoint + s0) * 3 + 1];
    float dA2 = xyz[((size_t)b0 * Npts + j0) * 3 + 2] - nxyz[((size_t)b0 * npoint + s0) * 3 + 2];
    float dB0 = xyz[((size_t)b1 * Npts + j1) * 3 + 0] - nxyz[((size_t)b1 * npoint + s1) * 3 + 0];
    float dB1 = xyz[((size_t)b1 * Npts + j1) * 3 + 1] - nxyz[((size_t)b1 * npoint + s1) * 3 + 1];
    float dB2 = xyz[((size_t)b1 * Npts + j1) * 3 + 2] - nxyz[((size_t)b1 * npoint + s1) * 3 + 2];

    const float *lp0, *lp1;                               // uniform-region base pointers
    if (LEVEL == 1) {
        lp0 = p1 + (size_t)b0 * 778 * N_ + j0;            // index (c-9)*N_
        lp1 = p1 + (size_t)b1 * 778 * N_ + j1;
    } else {
        lp0 = p0 + ((size_t)b0 * 256 + j0) * 128;         // index (c-3)
        lp1 = p0 + ((size_t)b1 * 256 + j1) * 128;
    }

    const float* Wc0 = W + n0;
    const float* Wc1 = W + n1;

    v8f acc00 = {}, acc01 = {}, acc10 = {}, acc11 = {};
    auto wstep = [&](v2f a0, v2f a1, v2f bb0, v2f bb1) {
        acc00 = __builtin_amdgcn_wmma_f32_16x16x4_f32(false, a0, false, bb0, (short)0, acc00, false, false);
        acc01 = __builtin_amdgcn_wmma_f32_16x16x4_f32(false, a0, false, bb1, (short)0, acc01, false, false);
        acc10 = __builtin_amdgcn_wmma_f32_16x16x4_f32(false, a1, false, bb0, (short)0, acc10, false, false);
        acc11 = __builtin_amdgcn_wmma_f32_16x16x4_f32(false, a1, false, bb1, (short)0, acc11, false, false);
    };
    auto gstep = [&](int kb) {                            // fully guarded step
        int ka = kb + 2 * half;
        v2f a0 = { sa_fetch<LEVEL>(ka,     K, b0, j0, p0, p1, p2, lp0, dA0, dA1, dA2),
                   sa_fetch<LEVEL>(ka + 1, K, b0, j0, p0, p1, p2, lp0, dA0, dA1, dA2) };
        v2f a1 = { sa_fetch<LEVEL>(ka,     K, b1, j1, p0, p1, p2, lp1, dB0, dB1, dB2),
                   sa_fetch<LEVEL>(ka + 1, K, b1, j1, p0, p1, p2, lp1, dB0, dB1, dB2) };
        v2f bb0 = { (ka < K) ? Wc0[(size_t)ka * Nc] : 0.f,
                    (ka + 1 < K) ? Wc0[(size_t)(ka + 1) * Nc] : 0.f };
        v2f bb1 = { (ka < K) ? Wc1[(size_t)ka * Nc] : 0.f,
                    (ka + 1 < K) ? Wc1[(size_t)(ka + 1) * Nc] : 0.f };
        wstep(a0, a1, bb0, bb1);
    };

    int kuni_lo = (LEVEL == 1) ? 12 : 4;
    int kuni_hi = (LEVEL == 1) ? 784 : (((K - 4) & ~3) + 4);

    for (int kb = 0; kb < kuni_lo; kb += 4) gstep(kb);
    for (int kb = kuni_lo; kb < kuni_hi; kb += 4) {       // branch-free hot loop
        int ka = kb + 2 * half;
        v2f a0, a1;
        if (LEVEL == 1) {
            a0 = v2f{ lp0[(size_t)(ka - 9) * N_], lp0[(size_t)(ka - 8) * N_] };
            a1 = v2f{ lp1[(size_t)(ka - 9) * N_], lp1[(size_t)(ka - 8) * N_] };
        } else {
            a0 = v2f{ lp0[ka - 3], lp0[ka - 2] };
            a1 = v2f{ lp1[ka - 3], lp1[ka - 2] };
        }
        v2f bb0 = { Wc0[(size_t)ka * Nc], Wc0[(size_t)(ka + 1) * Nc] };
        v2f bb1 = { Wc1[(size_t)ka * Nc], Wc1[(size_t)(ka + 1) * Nc] };
        wstep(a0, a1, bb0, bb1);
    }
    for (int kb = kuni_hi; kb < K; kb += 4) gstep(kb);

    float bi0 = bias[n0], bi1 = bias[n1];
    #pragma unroll
    for (int i = 0; i < 8; ++i) {
        int rr0 = tileM * 32 + i + 8 * half, rr1 = rr0 + 16;
        Y[(size_t)rr0 * Nc + n0] = acc00[i] + bi0;
        Y[(size_t)rr0 * Nc + n1] = acc01[i] + bi1;
        Y[(size_t)rr1 * Nc + n0] = acc10[i] + bi0;
        Y[(size_t)rr1 * Nc + n1] = acc11[i] + bi1;
    }
}

// ---------------------------------------------------------------------------
// BatchNorm (batch statistics, deterministic two-pass) + ReLU, maxpool
// ---------------------------------------------------------------------------
__global__ void bn_stats_kernel(const float* __restrict__ Y, float* __restrict__ mean,
                                float* __restrict__ var, int M, int Nc) {
    int c = blockIdx.x;
    __shared__ float ss[256], sq[256];
    float s = 0.f, q = 0.f;
    for (int r = threadIdx.x; r < M; r += 256) {
        float v = Y[(size_t)r * Nc + c];
        s += v; q += v * v;
    }
    ss[threadIdx.x] = s; sq[threadIdx.x] = q;
    __syncthreads();
    for (int st = 128; st > 0; st >>= 1) {
        if (threadIdx.x < st) {
            ss[threadIdx.x] += ss[threadIdx.x + st];
            sq[threadIdx.x] += sq[threadIdx.x + st];
        }
        __syncthreads();
    }
    if (threadIdx.x == 0) {
        float mm = ss[0] / (float)M;
        mean[c] = mm;
        var[c]  = sq[0] / (float)M - mm * mm;
    }
}

__global__ void bn_apply_relu_kernel(float* __restrict__ Y, const float* __restrict__ mean,
                                     const float* __restrict__ var, const float* __restrict__ g,
                                     const float* __restrict__ beta, size_t total, int Nc) {
    size_t i = (size_t)blockIdx.x * blockDim.x + threadIdx.x;
    if (i >= total) return;
    int c = (int)(i % (size_t)Nc);
    float v = (Y[i] - mean[c]) * rsqrtf(var[c] + EPS_BN) * g[c] + beta[c];
    Y[i] = v > 0.f ? v : 0.f;
}

__global__ void maxpool_kernel(const float* __restrict__ Y, float* __restrict__ out,
                               int G, int ns, int C) {
    int i = blockIdx.x * blockDim.x + threadIdx.x;       // over G*C
    if (i >= G * C) return;
    int gg = i / C, c = i - gg * C;
    float m = -3.4e38f;
    for (int t = 0; t < ns; ++t) m = fmaxf(m, Y[((size_t)gg * ns + t) * C + c]);
    out[i] = m;
}

// ---------------------------------------------------------------------------
// Concat / interpolation build kernels
// ---------------------------------------------------------------------------
__global__ void sa3_build_kernel(const float* __restrict__ l2xyz, const float* __restrict__ l2pts,
                                 float* __restrict__ out) {
    int i = blockIdx.x * blockDim.x + threadIdx.x;       // over 1024*259
    if (i >= 1024 * 259) return;
    int row = i / 259, c = i - row * 259;
    out[i] = (c < 3) ? l2xyz[(size_t)row * 3 + c] : l2pts[(size_t)row * 256 + (c - 3)];
}

__global__ void fc4_build_kernel(const float* __restrict__ xenc, const float* __restrict__ z,
                                 const float* __restrict__ cond, float* __restrict__ out) {
    int i = blockIdx.x * blockDim.x + threadIdx.x;       // over 8*592
    if (i >= 8 * 592) return;
    int b = i / 592, c = i - b * 592;
    float v;
    if (c < 512)      v = xenc[(size_t)b * 512 + c];
    else if (c < 528) v = z[(size_t)b * 16 + (c - 512)];
    else              v = cond[(size_t)b * 64 + (c - 528)];
    out[i] = v;
}

__global__ void fp3_build_kernel(const float* __restrict__ l2pts, const float* __restrict__ l3,
                                 float* __restrict__ out) {
    int i = blockIdx.x * blockDim.x + threadIdx.x;       // over 1024*768
    if (i >= 1024 * 768) return;
    int row = i / 768, c = i - row * 768;
    int b = row / 128;
    out[i] = (c < 256) ? l2pts[(size_t)row * 256 + c] : l3[(size_t)b * 512 + (c - 256)];
}

__global__ void fp2_build_kernel(const float* __restrict__ l1pts, const float* __restrict__ l2d,
                                 const int* __restrict__ idx, const float* __restrict__ w,
                                 float* __restrict__ out) {
    int i = blockIdx.x * blockDim.x + threadIdx.x;       // over 2048*384
    if (i >= 2048 * 384) return;
    int row = i / 384, c = i - row * 384;
    int b = row / 256;
    if (c < 128) { out[i] = l1pts[(size_t)row * 128 + c]; return; }
    int cc = c - 128;
    float v = 0.f;
    #pragma unroll
    for (int t = 0; t < 3; ++t)
        v += w[(size_t)row * 3 + t] * l2d[((size_t)b * 128 + idx[(size_t)row * 3 + t]) * 256 + cc];
    out[i] = v;
}

__global__ void fp1_build_kernel(const float* __restrict__ l0xyz, const float* __restrict__ feat,
                                 const float* __restrict__ l1d, const int* __restrict__ idx,
                                 const float* __restrict__ w, float* __restrict__ out) {
    size_t i = (size_t)blockIdx.x * blockDim.x + threadIdx.x;  // over 32768*137
    if (i >= (size_t)32768 * 137) return;
    int row = (int)(i / 137), c = (int)(i - (size_t)row * 137);
    int b = row / N_, n = row - b * N_;
    if (c < 3)      { out[i] = l0xyz[(size_t)row * 3 + c]; return; }
    if (c < 9)      { out[i] = feat[((size_t)b * 6 + (c - 3)) * N_ + n]; return; }
    int cc = c - 9;
    float v = 0.f;
    #pragma unroll
    for (int t = 0; t < 3; ++t)
        v += w[(size_t)row * 3 + t] * l1d[((size_t)b * 256 + idx[(size_t)row * 3 + t]) * 128 + cc];
    out[i] = v;
}

// Final: out[b, c, n] from xo (B*N,778) and cl (B*N) — LDS-tiled transpose
__global__ void out_transpose_kernel(const float* __restrict__ xo, const float* __restrict__ cl,
                                     float* __restrict__ out) {
    __shared__ float tile[32][33];
    int b = blockIdx.z;
    int n0 = blockIdx.x * 32, c0 = blockIdx.y * 32;
    int tx = threadIdx.x, ty = threadIdx.y;
    #pragma unroll
    for (int i = 0; i < 32; i += 8) {
        int n = n0 + ty + i, c = c0 + tx;
        float v = 0.f;
        if (c < 779) v = (c < 778) ? xo[((size_t)b * N_ + n) * 778 + c] : cl[(size_t)b * N_ + n];
        tile[ty + i][tx] = v;                       // tile[n_local][c_local]
    }
    __syncthreads();
    #pragma unroll
    for (int i = 0; i < 32; i += 8) {
        int c = c0 + ty + i, n = n0 + tx;
        if (c < 779) out[((size_t)b * 779 + c) * N_ + n] = tile[tx][ty + i];
    }
}

// ---------------------------------------------------------------------------
// Host side
// ---------------------------------------------------------------------------
extern "C" void kernel_launch(void* const* d_in, const int* in_sizes, int n_in,
                              void* d_out, int out_size, void* d_ws, size_t ws_size,
                              hipStream_t stream) {
    (void)n_in; (void)out_size; (void)ws_size;

    struct Layer { const float *w, *b, *g, *beta; };
    enum { SA1_0, SA1_1, SA2_0, SA2_1, SA3_0, SA3_1, FC4, FP3_0, FP3_1,
           FP2_0, FP2_1, FP1_0, FP1_1, CONV1, CONV2, CONTACT };
    Layer L[16];
    const float *contacts, *verts, *feat, *contactness, *zin, *cond;

    bool insertion = (in_sizes[0] == 8 * 778 * 4096);
    if (insertion) {
        contacts    = (const float*)d_in[0];
        verts       = (const float*)d_in[1];
        feat        = (const float*)d_in[2];
        contactness = (const float*)d_in[3];
        zin         = (const float*)d_in[4];
        cond        = (const float*)d_in[5];
        for (int li = 0; li < 16; ++li) {
            int base = 6 + 4 * li;   // order: w, b, g, beta
            L[li].w    = (const float*)d_in[base + 0];
            L[li].b    = (const float*)d_in[base + 1];
            L[li].g    = (const float*)d_in[base + 2];
            L[li].beta = (const float*)d_in[base + 3];
        }
    } else {
        // jax tree_flatten (sorted dict keys) layout
        cond        = (const float*)d_in[0];
        contactness = (const float*)d_in[1];
        contacts    = (const float*)d_in[2];
        feat        = (const float*)d_in[3];
        verts       = (const float*)d_in[68];
        zin         = (const float*)d_in[69];
        // sorted param bases (within layer: b, beta, g, w), logical order:
        const int bases[16] = {44, 48, 52, 56, 60, 64, 16, 36, 40, 28, 32, 20, 24, 8, 12, 4};
        for (int li = 0; li < 16; ++li) {
            int base = bases[li];
            L[li].b    = (const float*)d_in[base + 0];
            L[li].beta = (const float*)d_in[base + 1];
            L[li].g    = (const float*)d_in[base + 2];
            L[li].w    = (const float*)d_in[base + 3];
        }
    }

    float* ws = (float*)d_ws;
    float* l0xyz = ws + OF_L0XYZ;
    int*   fidx1 = (int*)(ws + OF_FIDX1);
    float* l1xyz = ws + OF_L1XYZ;
    int*   gi1   = (int*)(ws + OF_GI1);
    int*   fidx2 = (int*)(ws + OF_FIDX2);
    float* l2xyz = ws + OF_L2XYZ;
    int*   gi2   = (int*)(ws + OF_GI2);
    float* l1pts = ws + OF_L1PTS;
    float* l2pts = ws + OF_L2PTS;
    float* xenc  = ws + OF_XENC;
    float* fc4in = ws + OF_FC4IN;
    float* l3    = ws + OF_L3;
    float* l2d   = ws + OF_L2D;
    float* l1d   = ws + OF_L1D;
    int*   idx2  = (int*)(ws + OF_IDX2);
    float* w2    = ws + OF_W2;
    int*   idx1  = (int*)(ws + OF_IDX1);
    float* w1    = ws + OF_W1;
    float* meanb = ws + OF_MEAN;
    float* varb  = ws + OF_VAR;
    float* g3    = ws + OF_G3;
    float* fp3in = ws + OF_FP3IN;
    float* fp2in = ws + OF_FP2IN;
    float* fp1in = ws + OF_FP1IN;
    float* cl    = ws + OF_CL;
    float* bigA  = ws + OF_BIGA;
    float* bigB  = ws + OF_BIGB;
    float* xo    = ws + OF_XO;

    auto gemm = [&](const float* A, const Layer& ly, float* Y, int M, int K, int Nc) {
        int mt = (M + 31) / 32;
        dim3 grid((Nc + 31) / 32, (mt + 3) / 4);
        gemm_bias_kernel<<<grid, 128, 0, stream>>>(A, ly.w, ly.b, Y, M, K, Nc);
    };
    auto bn_relu = [&](float* Y, int M, int Nc, const Layer& ly) {
        bn_stats_kernel<<<Nc, 256, 0, stream>>>(Y, meanb, varb, M, Nc);
        size_t tot = (size_t)M * Nc;
        bn_apply_relu_kernel<<<(unsigned)((tot + 255) / 256), 256, 0, stream>>>(
            Y, meanb, varb, ly.g, ly.beta, tot, Nc);
    };

    // ---- l0 ----
    build_l0xyz_kernel<<<(B_ * N_ + 255) / 256, 256, 0, stream>>>(verts, l0xyz);

    // ---- SA1 ----
    fps_kernel<<<B_, 512, 0, stream>>>(l0xyz, fidx1, N_, 256);
    gather_xyz_kernel<<<(B_ * 256 + 255) / 256, 256, 0, stream>>>(l0xyz, fidx1, l1xyz, N_, 256);
    ball_kernel<<<(B_ * 256 + 255) / 256, 256, 0, stream>>>(l0xyz, l1xyz, gi1, N_, 256, 32, 0.04f);
    {
        dim3 grid(64 / 32, (65536 / 32) / 4);
        sa_gemm_kernel<1><<<grid, 128, 0, stream>>>(l0xyz, l1xyz, gi1, feat, contacts,
                                                    contactness, L[SA1_0].w, L[SA1_0].b,
                                                    bigA, N_, 256, 32, 788, 64);
    }
    bn_relu(bigA, 65536, 64, L[SA1_0]);
    gemm(bigA, L[SA1_1], bigB, 65536, 64, 128);
    bn_relu(bigB, 65536, 128, L[SA1_1]);
    maxpool_kernel<<<(2048 * 128 + 255) / 256, 256, 0, stream>>>(bigB, l1pts, 2048, 32, 128);

    // ---- SA2 ----
    fps_kernel<<<B_, 512, 0, stream>>>(l1xyz, fidx2, 256, 128);
    gather_xyz_kernel<<<(B_ * 128 + 255) / 256, 256, 0, stream>>>(l1xyz, fidx2, l2xyz, 256, 128);
    ball_kernel<<<(B_ * 128 + 255) / 256, 256, 0, stream>>>(l1xyz, l2xyz, gi2, 256, 128, 64, 0.0625f);
    {
        dim3 grid(128 / 32, (65536 / 32) / 4);
        sa_gemm_kernel<2><<<grid, 128, 0, stream>>>(l1xyz, l2xyz, gi2, l1pts, nullptr, nullptr,
                                                    L[SA2_0].w, L[SA2_0].b,
                                                    bigA, 256, 128, 64, 131, 128);
    }
    bn_relu(bigA, 65536, 128, L[SA2_0]);
    gemm(bigA, L[SA2_1], bigB, 65536, 128, 256);
    bn_relu(bigB, 65536, 256, L[SA2_1]);
    maxpool_kernel<<<(1024 * 256 + 255) / 256, 256, 0, stream>>>(bigB, l2pts, 1024, 64, 256);

    // ---- SA3 (group-all) ----
    sa3_build_kernel<<<(1024 * 259 + 255) / 256, 256, 0, stream>>>(l2xyz, l2pts, g3);
    gemm(g3, L[SA3_0], bigA, 1024, 259, 256);
    bn_relu(bigA, 1024, 256, L[SA3_0]);
    gemm(bigA, L[SA3_1], bigB, 1024, 256, 512);
    bn_relu(bigB, 1024, 512, L[SA3_1]);
    maxpool_kernel<<<(8 * 512 + 255) / 256, 256, 0, stream>>>(bigB, xenc, 8, 128, 512);

    // ---- FC4 ----
    fc4_build_kernel<<<(8 * 592 + 255) / 256, 256, 0, stream>>>(xenc, zin, cond, fc4in);
    gemm(fc4in, L[FC4], l3, 8, 592, 512);
    bn_relu(l3, 8, 512, L[FC4]);

    // ---- FP3 (S==1 broadcast) ----
    fp3_build_kernel<<<(1024 * 768 + 255) / 256, 256, 0, stream>>>(l2pts, l3, fp3in);
    gemm(fp3in, L[FP3_0], bigA, 1024, 768, 512);
    bn_relu(bigA, 1024, 512, L[FP3_0]);
    gemm(bigA, L[FP3_1], l2d, 1024, 512, 256);
    bn_relu(l2d, 1024, 256, L[FP3_1]);

    // ---- FP2 ----
    knn3_kernel<<<(B_ * 256 + 255) / 256, 256, 0, stream>>>(l1xyz, l2xyz, 256, 128, idx2, w2);
    fp2_build_kernel<<<(2048 * 384 + 255) / 256, 256, 0, stream>>>(l1pts, l2d, idx2, w2, fp2in);
    gemm(fp2in, L[FP2_0], bigA, 2048, 384, 256);
    bn_relu(bigA, 2048, 256, L[FP2_0]);
    gemm(bigA, L[FP2_1], l1d, 2048, 256, 128);
    bn_relu(l1d, 2048, 128, L[FP2_1]);

    // ---- FP1 ----
    knn3_kernel<<<(B_ * N_ + 255) / 256, 256, 0, stream>>>(l0xyz, l1xyz, N_, 256, idx1, w1);
    {
        size_t tot = (size_t)32768 * 137;
        fp1_build_kernel<<<(unsigned)((tot + 255) / 256), 256, 0, stream>>>(
            l0xyz, feat, l1d, idx1, w1, fp1in);
    }
    gemm(fp1in, L[FP1_0], bigA, 32768, 137, 128);
    bn_relu(bigA, 32768, 128, L[FP1_0]);
    gemm(bigA, L[FP1_1], bigB, 32768, 128, 128);          // l0d -> bigB
    bn_relu(bigB, 32768, 128, L[FP1_1]);

    // ---- heads ----
    gemm(bigB, L[CONV1], bigA, 32768, 128, 128);          // feat -> bigA
    bn_relu(bigA, 32768, 128, L[CONV1]);
    gemm(bigA, L[CONV2], xo, 32768, 128, 778);            // bias only, no BN
    gemm(bigA, L[CONTACT], cl, 32768, 128, 1);            // bias only, no BN

    out_transpose_kernel<<<dim3(N_ / 32, (779 + 31) / 32, B_), dim3(32, 8), 0, stream>>>(
        xo, cl, (float*)d_out);
}